// QuantumLayer_MultiQunode_67010079752483
// MI455X (gfx1250) — compile-verified
//
#include <hip/hip_runtime.h>
#include <math.h>

// ---------------------------------------------------------------------------
// CV quantum circuit simulator for MI455X (gfx1250).
//   Phase 1: 464 small (<=14x14) matrix exponentials (all gates; BS is block-
//            diagonal in total photon number -> 27 blocks of <=14).
//   Phase 2: compose per-circuit 196x196 unitary U[c] via structured applies.
//   Phase 3: per-batch coherent-encoding states via Taylor on the truncated
//            tridiagonal displacement generator (matches reference expm).
//   Phase 4: batched complex GEMM (8192x196)x(196x196) with fp32 WMMA
//            (v_wmma_f32_16x16x4_f32), fused <x> expectation epilogue in LDS.
//            SoA re/im planes -> shuffle-free fragment loads; 16 j-tiles so
//            each of the 4 waves owns exactly 4 tiles (no EXEC masking).
// ---------------------------------------------------------------------------

#define CUT 14
#define DIM2 196           // CUT^2
#define NCIRC 2
#define NLAYER 4
#define NPAD 256           // 16*16: j padded so tiles split 4/4/4/4 over waves
#define SLOTS_PER_CL 58    // 27 BS1 blocks + 27 BS2 blocks + 2 squeeze + 2 disp
#define NTASK (NCIRC * NLAYER * SLOTS_PER_CL)  // 464

typedef __attribute__((ext_vector_type(2))) float v2f;
typedef __attribute__((ext_vector_type(8))) float v8f;

enum { OP_BS = 0, OP_MODE0 = 1, OP_MODE1 = 2, OP_ROT = 3, OP_KERR = 4 };

static __device__ inline v8f wmma4(v2f a, v2f b, v8f c) {
  // D(16x16,f32) += A(16x4,f32) * B(4x16,f32); f32 path has no A/B NEG.
  return __builtin_amdgcn_wmma_f32_16x16x4_f32(false, a, false, b,
                                               (short)0, c, false, false);
}

// ---------------------------------------------------------------------------
// Phase 1: generic dense expm (dim padded to 14) in one workgroup per task.
// ---------------------------------------------------------------------------
__global__ __launch_bounds__(256) void expm_tasks(
    const float* __restrict__ th1, const float* __restrict__ p1,
    const float* __restrict__ th2, const float* __restrict__ p2,
    const float* __restrict__ rr,  const float* __restrict__ phr,
    const float* __restrict__ aa,  const float* __restrict__ pha,
    float2* __restrict__ slots) {
  __shared__ float2 Xm[CUT][16], Rm[CUT][16], Tm[CUT][16];
  __shared__ int ssh;
  const int task = blockIdx.x;
  const int cl = task / SLOTS_PER_CL;
  const int rem = task % SLOTS_PER_CL;
  const int tid = threadIdx.x;
  const int row = tid / CUT, col = tid % CUT;
  const bool act = tid < DIM2;

  if (act) Xm[row][col] = make_float2(0.f, 0.f);
  __syncthreads();

  if (tid == 0) {
    if (rem < 54) {                 // beamsplitter block, total photons N
      const int N = (rem < 27) ? rem : rem - 27;
      const float th = (rem < 27) ? th1[cl] : th2[cl];
      const float ph = (rem < 27) ? p1[cl] : p2[cl];
      float sp, cp; sincosf(ph, &sp, &cp);
      const float cr = th * cp, ci = th * sp;
      const int lo = max(0, N - (CUT - 1));
      const int hi = min(CUT - 1, N);
      const int d = hi - lo + 1;
      for (int i = 0; i + 1 < d; ++i) {
        const int n0 = lo + i, n1 = N - n0;
        const float g = sqrtf((float)(n0 + 1)) * sqrtf((float)n1);
        Xm[i + 1][i] = make_float2(cr * g, ci * g);     //  th e^{+i ph} g
        Xm[i][i + 1] = make_float2(-cr * g, ci * g);    // -th e^{-i ph} g
      }
    } else if (rem < 56) {          // squeeze S(z), z = r e^{i ph}
      const int m = rem - 54;
      const float r = rr[cl * 2 + m], ph = phr[cl * 2 + m];
      float sp, cp; sincosf(ph, &sp, &cp);
      const float zr = r * cp, zi = r * sp;
      for (int n = 0; n + 2 < CUT; ++n) {
        const float g = 0.5f * sqrtf((float)((n + 1) * (n + 2)));
        Xm[n][n + 2] = make_float2(zr * g, -zi * g);    //  0.5 conj(z) g
        Xm[n + 2][n] = make_float2(-zr * g, -zi * g);   // -0.5 z g
      }
    } else {                        // displacement D(alpha)
      const int m = rem - 56;
      const float a = aa[cl * 2 + m], ph = pha[cl * 2 + m];
      float sp, cp; sincosf(ph, &sp, &cp);
      const float ar = a * cp, ai = a * sp;
      for (int n = 0; n + 1 < CUT; ++n) {
        const float g = sqrtf((float)(n + 1));
        Xm[n + 1][n] = make_float2(ar * g, ai * g);     //  alpha g
        Xm[n][n + 1] = make_float2(-ar * g, ai * g);    // -conj(alpha) g
      }
    }
  }
  __syncthreads();

  if (tid == 0) {                   // 1-norm -> scaling exponent
    float mx = 0.f;
    for (int c2 = 0; c2 < CUT; ++c2) {
      float s = 0.f;
      for (int r2 = 0; r2 < CUT; ++r2)
        s += fabsf(Xm[r2][c2].x) + fabsf(Xm[r2][c2].y);
      mx = fmaxf(mx, s);
    }
    int s = 0;
    while (mx > 0.5f && s < 24) { mx *= 0.5f; ++s; }
    ssh = s;
  }
  __syncthreads();
  const int sexp = ssh;
  if (act) {
    const float sc = ldexpf(1.f, -sexp);
    Xm[row][col].x *= sc; Xm[row][col].y *= sc;
    Rm[row][col] = make_float2(row == col ? 1.f : 0.f, 0.f);
  }
  __syncthreads();

  // Horner Taylor-16: R <- I + (X*R)/k, k = 16..1
  for (int k = 16; k >= 1; --k) {
    if (act) {
      float tr = 0.f, ti = 0.f;
      for (int m2 = 0; m2 < CUT; ++m2) {
        const float2 xa = Xm[row][m2], rb = Rm[m2][col];
        tr += xa.x * rb.x - xa.y * rb.y;
        ti += xa.x * rb.y + xa.y * rb.x;
      }
      Tm[row][col] = make_float2(tr, ti);
    }
    __syncthreads();
    if (act) {
      const float inv = 1.f / (float)k;
      Rm[row][col] = make_float2((row == col ? 1.f : 0.f) + Tm[row][col].x * inv,
                                 Tm[row][col].y * inv);
    }
    __syncthreads();
  }
  // squaring
  for (int q = 0; q < sexp; ++q) {
    if (act) {
      float tr = 0.f, ti = 0.f;
      for (int m2 = 0; m2 < CUT; ++m2) {
        const float2 xa = Rm[row][m2], rb = Rm[m2][col];
        tr += xa.x * rb.x - xa.y * rb.y;
        ti += xa.x * rb.y + xa.y * rb.x;
      }
      Tm[row][col] = make_float2(tr, ti);
    }
    __syncthreads();
    if (act) Rm[row][col] = Tm[row][col];
    __syncthreads();
  }
  if (act) slots[(size_t)task * DIM2 + row * CUT + col] = Rm[row][col];
}

// ---------------------------------------------------------------------------
// Phase 2: structured 196x196 applies, out = Op * in (full rewrite each call).
// ---------------------------------------------------------------------------
__global__ void set_id(float2* __restrict__ u) {
  const int t = blockIdx.x * blockDim.x + threadIdx.x;
  if (t < DIM2 * DIM2)
    u[t] = make_float2((t / DIM2) == (t % DIM2) ? 1.f : 0.f, 0.f);
}

__global__ void apply_op(float2* __restrict__ out, const float2* __restrict__ in,
                         const float2* __restrict__ slot,
                         const float* __restrict__ par, int cl, int op) {
  const int t = blockIdx.x * blockDim.x + threadIdx.x;
  if (t >= DIM2 * DIM2) return;
  const int j = t / DIM2, col = t % DIM2;
  const int n0 = j / CUT, n1 = j % CUT;
  float accx = 0.f, accy = 0.f;

  if (op == OP_BS) {
    const int N = n0 + n1;
    const int lo = max(0, N - (CUT - 1));
    const int hi = min(CUT - 1, N);
    const int d = hi - lo + 1;
    const int i = n0 - lo;
    const float2* blk = slot + (size_t)N * DIM2;
    for (int ip = 0; ip < d; ++ip) {
      const float2 m = blk[i * CUT + ip];
      const int n0p = lo + ip;
      const float2 u = in[(size_t)(n0p * CUT + (N - n0p)) * DIM2 + col];
      accx += m.x * u.x - m.y * u.y;
      accy += m.x * u.y + m.y * u.x;
    }
  } else if (op == OP_MODE0) {
    for (int m2 = 0; m2 < CUT; ++m2) {
      const float2 a = slot[n0 * CUT + m2];
      const float2 u = in[(size_t)(m2 * CUT + n1) * DIM2 + col];
      accx += a.x * u.x - a.y * u.y;
      accy += a.x * u.y + a.y * u.x;
    }
  } else if (op == OP_MODE1) {
    for (int m2 = 0; m2 < CUT; ++m2) {
      const float2 a = slot[n1 * CUT + m2];
      const float2 u = in[(size_t)(n0 * CUT + m2) * DIM2 + col];
      accx += a.x * u.x - a.y * u.y;
      accy += a.x * u.y + a.y * u.x;
    }
  } else {
    float phase;
    if (op == OP_ROT)
      phase = par[cl * 2 + 0] * (float)n0 + par[cl * 2 + 1] * (float)n1;
    else
      phase = par[cl * 2 + 0] * (float)(n0 * n0) +
              par[cl * 2 + 1] * (float)(n1 * n1);
    float sp, cp; sincosf(phase, &sp, &cp);
    const float2 u = in[(size_t)j * DIM2 + col];
    accx = cp * u.x - sp * u.y;
    accy = cp * u.y + sp * u.x;
  }
  out[(size_t)j * DIM2 + col] = make_float2(accx, accy);
}

// Pack final U into SoA planes: upR/upI are [NPAD][DIM2] with zero rows >=196.
__global__ void pack_u(float* __restrict__ upR, float* __restrict__ upI,
                       const float2* __restrict__ u) {
  const int t = blockIdx.x * blockDim.x + threadIdx.x;
  if (t >= NPAD * DIM2) return;
  const int j = t / DIM2, k = t % DIM2;
  const float2 v = (j < DIM2) ? u[(size_t)j * DIM2 + k] : make_float2(0.f, 0.f);
  upR[t] = v.x;
  upI[t] = v.y;
}

// ---------------------------------------------------------------------------
// Phase 3: encoding states (SoA re/im planes).
// ---------------------------------------------------------------------------
static __device__ void coherent_trunc(float mag, float ph, float2* v) {
  float sp, cp; sincosf(ph, &sp, &cp);
  const float ar = mag * cp, ai = mag * sp;
  float2 term[CUT];
#pragma unroll
  for (int n = 0; n < CUT; ++n) {
    term[n] = make_float2(0.f, 0.f);
    v[n] = make_float2(0.f, 0.f);
  }
  term[0].x = 1.f; v[0].x = 1.f;
  for (int k = 1; k <= 60; ++k) {
    const float inv = 1.f / (float)k;
    float2 nt[CUT];
#pragma unroll
    for (int n = 0; n < CUT; ++n) {
      float sx = 0.f, sy = 0.f;
      if (n >= 1) {                     //  alpha sqrt(n) term[n-1]
        const float g = sqrtf((float)n);
        sx += (ar * term[n - 1].x - ai * term[n - 1].y) * g;
        sy += (ar * term[n - 1].y + ai * term[n - 1].x) * g;
      }
      if (n <= CUT - 2) {               // -conj(alpha) sqrt(n+1) term[n+1]
        const float g = sqrtf((float)(n + 1));
        sx -= (ar * term[n + 1].x + ai * term[n + 1].y) * g;
        sy -= (ar * term[n + 1].y - ai * term[n + 1].x) * g;
      }
      nt[n] = make_float2(sx * inv, sy * inv);
    }
#pragma unroll
    for (int n = 0; n < CUT; ++n) {
      term[n] = nt[n];
      v[n].x += nt[n].x; v[n].y += nt[n].y;
    }
  }
}

// initP layout: plane-major [plane][c][b][k]; plane stride = NCIRC*Bn*DIM2.
__global__ void init_states(const float* __restrict__ x,
                            float* __restrict__ initP, int Bn) {
  const int t = blockIdx.x * blockDim.x + threadIdx.x;
  if (t >= Bn * NCIRC) return;
  const int c = t / Bn, b = t % Bn;
  const float* f = x + (size_t)b * (NCIRC * 4) + c * 4;  // [mag0,mag1,ph0,ph1]
  float2 d0[CUT], d1[CUT];
  coherent_trunc(f[0], f[2], d0);
  coherent_trunc(f[1], f[3], d1);
  const size_t ps = (size_t)NCIRC * Bn * DIM2;
  float* oR = initP + ((size_t)c * Bn + b) * DIM2;
  float* oI = oR + ps;
#pragma unroll
  for (int n0 = 0; n0 < CUT; ++n0)
#pragma unroll
    for (int n1 = 0; n1 < CUT; ++n1) {
      oR[n0 * CUT + n1] = d0[n0].x * d1[n1].x - d0[n0].y * d1[n1].y;
      oI[n0 * CUT + n1] = d0[n0].x * d1[n1].y + d0[n0].y * d1[n1].x;
    }
}

// ---------------------------------------------------------------------------
// Phase 4: fused complex GEMM (WMMA fp32) + <x> expectation epilogue.
// Block: 128 threads = 4 waves; 16 batch rows; 16 j-tiles, 4 fixed per wave.
// ---------------------------------------------------------------------------
__global__ __launch_bounds__(128) void gemm_expect(
    const float* __restrict__ initP, const float* __restrict__ Upk,
    float* __restrict__ out, int Bn) {
  const int c = blockIdx.y;
  const int row0 = blockIdx.x * 16;
  const int lane = threadIdx.x & 31;
  const int wave = threadIdx.x >> 5;
  const int m16 = lane & 15;
  const int khi = (lane >> 4) << 1;   // 0 or 2: K sub-offset for this half-wave

  const size_t ps = (size_t)NCIRC * Bn * DIM2;          // init plane stride
  const float* aR = initP + ((size_t)c * Bn + row0 + m16) * DIM2;
  const float* aI = aR + ps;
  const float* uR = Upk + (size_t)c * 2 * NPAD * DIM2;  // [c][plane][j][k]
  const float* uI = uR + (size_t)NPAD * DIM2;

  // fixed 4 tiles per wave: wave, wave+4, wave+8, wave+12 (rows >=196 are 0)
  size_t boff[4];
#pragma unroll
  for (int i = 0; i < 4; ++i)
    boff[i] = (size_t)((wave + 4 * i) * 16 + m16) * DIM2;

  const v8f vzero = {0.f, 0.f, 0.f, 0.f, 0.f, 0.f, 0.f, 0.f};
  v8f accR[4], accI[4];
#pragma unroll
  for (int i = 0; i < 4; ++i) { accR[i] = vzero; accI[i] = vzero; }

  for (int k0 = 0; k0 < DIM2; k0 += 4) {
    const int kk = k0 + khi;
    const v2f ar = *(const v2f*)(aR + kk);
    const v2f ai = *(const v2f*)(aI + kk);
    const v2f an = -ai;
#pragma unroll
    for (int i = 0; i < 4; ++i) {
      const v2f br = *(const v2f*)(uR + boff[i] + kk);
      const v2f bi = *(const v2f*)(uI + boff[i] + kk);
      accR[i] = wmma4(ar, br, accR[i]);   // Re*Re
      accR[i] = wmma4(an, bi, accR[i]);   // -Im*Im
      accI[i] = wmma4(ar, bi, accI[i]);   // Re*Im
      accI[i] = wmma4(ai, br, accI[i]);   // Im*Re
    }
  }

  __shared__ float2 st[16][DIM2];
  const int mrow = (lane >> 4) * 8;    // D layout: vgpr r -> M=r (+8 hi half)
#pragma unroll
  for (int i = 0; i < 4; ++i) {
    const int j = (wave + 4 * i) * 16 + m16;
    if (j < DIM2) {
#pragma unroll
      for (int r = 0; r < 8; ++r)
        st[mrow + r][j] = make_float2(accR[i][r], accI[i][r]);
    }
  }
  __syncthreads();

  if (threadIdx.x < 32) {
    const int row = threadIdx.x & 15;
    const int mode = threadIdx.x >> 4;
    float e = 0.f;
    if (mode == 0) {
      for (int n0 = 1; n0 < CUT; ++n0) {
        const float s = 2.f * sqrtf((float)n0);
        for (int n1 = 0; n1 < CUT; ++n1) {
          const float2 p = st[row][n0 * CUT + n1];
          const float2 q = st[row][(n0 - 1) * CUT + n1];
          e += s * (p.x * q.x + p.y * q.y);
        }
      }
    } else {
      for (int n1 = 1; n1 < CUT; ++n1) {
        const float s = 2.f * sqrtf((float)n1);
        for (int n0 = 0; n0 < CUT; ++n0) {
          const float2 p = st[row][n0 * CUT + n1];
          const float2 q = st[row][n0 * CUT + n1 - 1];
          e += s * (p.x * q.x + p.y * q.y);
        }
      }
    }
    out[(size_t)(row0 + row) * (NCIRC * 2) + c * 2 + mode] = e;
  }
}

// ---------------------------------------------------------------------------
extern "C" void kernel_launch(void* const* d_in, const int* in_sizes, int n_in,
                              void* d_out, int out_size, void* d_ws,
                              size_t ws_size, hipStream_t stream) {
  const float* x   = (const float*)d_in[0];
  const float* th1 = (const float*)d_in[1];
  const float* p1  = (const float*)d_in[2];
  const float* vp1 = (const float*)d_in[3];
  const float* rr  = (const float*)d_in[4];
  const float* phr = (const float*)d_in[5];
  const float* th2 = (const float*)d_in[6];
  const float* p2  = (const float*)d_in[7];
  const float* vp2 = (const float*)d_in[8];
  const float* aa  = (const float*)d_in[9];
  const float* pha = (const float*)d_in[10];
  const float* kk  = (const float*)d_in[11];
  const int Bn = in_sizes[0] / (NCIRC * 4);

  float* wsf = (float*)d_ws;
  size_t off = 0;
  float2* slots = (float2*)(wsf + off); off += (size_t)2 * NTASK * DIM2;
  float2* Ua    = (float2*)(wsf + off); off += (size_t)2 * DIM2 * DIM2;
  float2* Ub    = (float2*)(wsf + off); off += (size_t)2 * DIM2 * DIM2;
  float*  Upk   = wsf + off;            off += (size_t)NCIRC * 2 * NPAD * DIM2;
  float*  initP = wsf + off;            off += (size_t)2 * NCIRC * Bn * DIM2;
  (void)ws_size; (void)n_in; (void)out_size;

  expm_tasks<<<NTASK, 256, 0, stream>>>(th1, p1, th2, p2, rr, phr, aa, pha,
                                        slots);
  init_states<<<(Bn * NCIRC + 127) / 128, 128, 0, stream>>>(x, initP, Bn);

  const int GRID_A = (DIM2 * DIM2 + 255) / 256;
  for (int c = 0; c < NCIRC; ++c) {
    set_id<<<GRID_A, 256, 0, stream>>>(Ua);
    float2* cur = Ua;
    float2* alt = Ub;
#define STEP(SLOT, PAR, OP)                                                  \
    do {                                                                     \
      apply_op<<<GRID_A, 256, 0, stream>>>(alt, cur, SLOT, PAR, cl, OP);     \
      float2* tsw = cur; cur = alt; alt = tsw;                               \
    } while (0)
    for (int l = 0; l < NLAYER; ++l) {
      const int cl = c * NLAYER + l;
      const float2* base = slots + (size_t)cl * SLOTS_PER_CL * DIM2;
      STEP(base,                       nullptr, OP_BS);     // BS1 (27 blocks)
      STEP(nullptr,                    vp1,     OP_ROT);    // varphi_1 diag
      STEP(base + (size_t)54 * DIM2,   nullptr, OP_MODE0);  // squeeze mode 0
      STEP(base + (size_t)55 * DIM2,   nullptr, OP_MODE1);  // squeeze mode 1
      STEP(base + (size_t)27 * DIM2,   nullptr, OP_BS);     // BS2
      STEP(nullptr,                    vp2,     OP_ROT);    // varphi_2 diag
      STEP(base + (size_t)56 * DIM2,   nullptr, OP_MODE0);  // disp mode 0
      STEP(base + (size_t)57 * DIM2,   nullptr, OP_MODE1);  // disp mode 1
      STEP(nullptr,                    kk,      OP_KERR);   // Kerr diag
    }
#undef STEP
    pack_u<<<(NPAD * DIM2 + 255) / 256, 256, 0, stream>>>(
        Upk + (size_t)c * 2 * NPAD * DIM2,
        Upk + (size_t)c * 2 * NPAD * DIM2 + (size_t)NPAD * DIM2, cur);
  }

  dim3 grid(Bn / 16, NCIRC);
  gemm_expect<<<grid, 128, 0, stream>>>(initP, Upk, (float*)d_out, Bn);
}